// CombinedHiddenGCAE_16286515987228
// MI455X (gfx1250) — compile-verified
//
#include <hip/hip_runtime.h>
#include <math.h>

typedef float v2f __attribute__((ext_vector_type(2)));
typedef float v8f __attribute__((ext_vector_type(8)));

#define NN 100000
#define NE 1600000

// ---------------------------------------------------------------------------
// GEMM  H[M,NOUT] = X[M,K] @ W[K,NOUT]   via V_WMMA_F32_16X16X4_F32.
// W is staged in LDS once per workgroup (<=80KB; 4 blocks / 320KB WGP), with
// consecutive k-rows interleaved so each B fragment is one ds_load_b64.
// One wave owns a 16-row strip; keeps NOUT/16 (<=8) v8f accumulators live,
// reusing the A fragment across all column tiles per K-step.
// f32 16x16x4 lane layout (ISA 7.12.2):
//   A: lane l -> row (l&15); vgpr0/1 hold K = (l>>4)*2 + {0,1}
//   B: lane l -> col (l&15); vgpr0/1 hold K = (l>>4)*2 + {0,1}
//   C/D: vgpr r -> row r + 8*(l>>4), col (l&15)
// ---------------------------------------------------------------------------
template <int K, int NOUT>
__global__ __launch_bounds__(256) void gemm_wmma(const float* __restrict__ X,
                                                 const float* __restrict__ W,
                                                 float* __restrict__ H, int M) {
  constexpr int NT = NOUT / 16;
  // layout: Wlds[kpair][n][2]  where kpair = krow>>1, last index = krow&1
  __shared__ float Wlds[(K / 2) * NOUT * 2];

  for (int e = threadIdx.x; e < K * NOUT; e += 256) {
    int krow = e / NOUT, n = e - krow * NOUT;          // coalesced global read
    Wlds[(krow >> 1) * (NOUT * 2) + n * 2 + (krow & 1)] = W[e];
  }
  __syncthreads();

  const int lane   = threadIdx.x & 31;
  const int wave   = threadIdx.x >> 5;
  const int strip  = blockIdx.x * (blockDim.x >> 5) + wave;
  const int row0   = strip * 16;
  if (row0 >= M) return;  // uniform per wave -> EXEC stays all-ones for WMMA

  const int lanehi = lane >> 4;   // 0 or 1
  const int lanelo = lane & 15;
  const int kb     = lanehi * 2;  // K sub-offset handled by this half-wave

  v8f acc[NT];
#pragma unroll
  for (int t = 0; t < NT; ++t) acc[t] = (v8f){0.f,0.f,0.f,0.f,0.f,0.f,0.f,0.f};

  const float* xrow = X + (size_t)(row0 + lanelo) * K;
#pragma unroll 4
  for (int k = 0; k < K; k += 4) {
    v2f a = *(const v2f*)(xrow + k + kb);                      // global_load_b64
    const float* wrow = &Wlds[((k + kb) >> 1) * (NOUT * 2)];
#pragma unroll
    for (int t = 0; t < NT; ++t) {
      v2f b = *(const v2f*)(wrow + (t * 16 + lanelo) * 2);     // ds_load_b64
      acc[t] = __builtin_amdgcn_wmma_f32_16x16x4_f32(
          /*neg_a=*/false, a, /*neg_b=*/false, b,
          /*c_mod=*/(short)0, acc[t], /*reuse_a=*/false, /*reuse_b=*/false);
    }
  }

  const int rbase = row0 + 8 * lanehi;
#pragma unroll
  for (int t = 0; t < NT; ++t)
#pragma unroll
    for (int r = 0; r < 8; ++r)
      H[(size_t)(rbase + r) * NOUT + t * 16 + lanelo] = acc[t][r];
}

// ---------------------------------------------------------------------------
// Degree / normalization
// ---------------------------------------------------------------------------
__global__ void fill_zero_k(float* p, unsigned n) {
  unsigned i = blockIdx.x * blockDim.x + threadIdx.x;
  if (i < n) p[i] = 0.f;
}

__global__ void degree_k(const long long* __restrict__ dst, float* deg, int E) {
  int e = blockIdx.x * blockDim.x + threadIdx.x;
  if (e < E) unsafeAtomicAdd(&deg[(int)dst[e]], 1.0f);
}

__global__ void dinv_k(const float* __restrict__ deg, float* dinv, int n) {
  int i = blockIdx.x * blockDim.x + threadIdx.x;
  if (i < n) dinv[i] = rsqrtf(deg[i] + 1.0f);  // self-loop adds 1
}

// ---------------------------------------------------------------------------
// agg = h * dinv^2 + bias   (self-loop term + bias, also zero-inits agg)
// ---------------------------------------------------------------------------
template <int F>
__global__ void agg_init_k(const float* __restrict__ h, const float* __restrict__ dinv,
                           const float* __restrict__ bias, float* __restrict__ agg,
                           int n) {
  constexpr unsigned C = F / 4;
  unsigned tid = blockIdx.x * blockDim.x + threadIdx.x;
  if (tid >= (unsigned)n * C) return;
  unsigned node = tid / C, c = tid % C;
  float di = dinv[node];
  float w  = di * di;
  float4 hv = *(const float4*)(h + (size_t)node * F + c * 4);
  float4 bv = *(const float4*)(bias + c * 4);
  float4 o;
  o.x = hv.x * w + bv.x;  o.y = hv.y * w + bv.y;
  o.z = hv.z * w + bv.z;  o.w = hv.w * w + bv.w;
  *(float4*)(agg + (size_t)node * F + c * 4) = o;
}

// ---------------------------------------------------------------------------
// Edge scatter: agg[dst] += h[src] * dinv[src]*dinv[dst].
// Thread handles one float4 chunk; a wave covers one full 128-wide row ->
// coalesced gather + coalesced no-return f32 atomics (L2-resident, 192MB L2).
// ---------------------------------------------------------------------------
template <int F>
__global__ void edge_scatter_k(const float* __restrict__ h,
                               const long long* __restrict__ src,
                               const long long* __restrict__ dst,
                               const float* __restrict__ dinv,
                               float* __restrict__ agg, int E) {
  constexpr unsigned C = F / 4;
  unsigned tid = blockIdx.x * blockDim.x + threadIdx.x;
  if (tid >= (unsigned)E * C) return;   // E*C = 51.2M < 2^32
  unsigned e = tid / C, c = tid % C;
  int s = (int)src[e], d = (int)dst[e];
  float coef = dinv[s] * dinv[d];
  float4 hv = *(const float4*)(h + (size_t)s * F + c * 4);
  float* out = agg + (size_t)d * F + c * 4;
  unsafeAtomicAdd(out + 0, hv.x * coef);
  unsafeAtomicAdd(out + 1, hv.y * coef);
  unsafeAtomicAdd(out + 2, hv.z * coef);
  unsafeAtomicAdd(out + 3, hv.w * coef);
}

__global__ void tanh_k(float* p, unsigned n) {
  unsigned i = blockIdx.x * blockDim.x + threadIdx.x;
  if (i < n) p[i] = tanhf(p[i]);
}

// ---------------------------------------------------------------------------
// Concats:  [feature(128)|cond(32)] -> 160 cols ;  [z(64)|cond(32)] -> 96 cols
// ---------------------------------------------------------------------------
__global__ void concat_fc_k(const float* __restrict__ feat, const float* __restrict__ cond,
                            float* __restrict__ out, int n) {
  unsigned tid = blockIdx.x * blockDim.x + threadIdx.x;
  if (tid >= (unsigned)n * 40u) return;
  unsigned node = tid / 40u, c = tid % 40u;
  float4 v = (c < 32u) ? *(const float4*)(feat + (size_t)node * 128 + c * 4)
                       : *(const float4*)(cond + (size_t)node * 32 + (c - 32u) * 4);
  *(float4*)(out + (size_t)node * 160 + c * 4) = v;
}

__global__ void concat_zc_k(const float* __restrict__ z, const float* __restrict__ cond,
                            float* __restrict__ out, int n) {
  unsigned tid = blockIdx.x * blockDim.x + threadIdx.x;
  if (tid >= (unsigned)n * 24u) return;
  unsigned node = tid / 24u, c = tid % 24u;
  float4 v = (c < 16u) ? *(const float4*)(z + (size_t)node * 64 + c * 4)
                       : *(const float4*)(cond + (size_t)node * 32 + (c - 16u) * 4);
  *(float4*)(out + (size_t)node * 96 + c * 4) = v;
}

// ---------------------------------------------------------------------------
static inline int cdiv(long long a, int b) { return (int)((a + b - 1) / b); }

extern "C" void kernel_launch(void* const* d_in, const int* in_sizes, int n_in,
                              void* d_out, int out_size, void* d_ws, size_t ws_size,
                              hipStream_t stream) {
  const float* feature   = (const float*)d_in[0];
  const float* condition = (const float*)d_in[1];
  const long long* eidx  = (const long long*)d_in[2];  // int64 [2,E]
  const float* W_e1 = (const float*)d_in[3];  const float* b_e1 = (const float*)d_in[4];
  const float* W_e2 = (const float*)d_in[5];  const float* b_e2 = (const float*)d_in[6];
  const float* W_e3 = (const float*)d_in[7];  const float* b_e3 = (const float*)d_in[8];
  const float* W_d1 = (const float*)d_in[9];  const float* b_d1 = (const float*)d_in[10];
  const float* W_d2 = (const float*)d_in[11]; const float* b_d2 = (const float*)d_in[12];
  const float* W_d3 = (const float*)d_in[13]; const float* b_d3 = (const float*)d_in[14];

  const long long* src = eidx;        // row 0
  const long long* dst = eidx + NE;   // row 1

  // workspace layout (floats): deg[N] dinv[N] IN[N*160] GM[N*128] AG[N*128]
  float* ws   = (float*)d_ws;
  float* deg  = ws;
  float* dinv = deg + NN;
  float* IN   = dinv + NN;
  float* GM   = IN + (size_t)NN * 160;
  float* AG   = GM + (size_t)NN * 128;
  float* OUT  = (float*)d_out;        // [N,128]

  const int B = 256;
  const int gemmGrid = cdiv(NN / 16, 8);           // 8 waves (strips) per block
  const int nodeF128 = cdiv((long long)NN * 32, B);
  const int nodeF64  = cdiv((long long)NN * 16, B);
  const int edgeF128 = cdiv((long long)NE * 32, B);
  const int edgeF64  = cdiv((long long)NE * 16, B);
  const int elemF128 = cdiv((long long)NN * 128, B);

  // degree + normalization
  fill_zero_k<<<cdiv(NN, B), B, 0, stream>>>(deg, NN);
  degree_k<<<cdiv(NE, B), B, 0, stream>>>(dst, deg, NE);
  dinv_k<<<cdiv(NN, B), B, 0, stream>>>(deg, dinv, NN);

  // encoder input concat
  concat_fc_k<<<cdiv((long long)NN * 40, B), B, 0, stream>>>(feature, condition, IN, NN);

  // ---- encoder layer 1: 160 -> 128, tanh ----
  gemm_wmma<160, 128><<<gemmGrid, B, 0, stream>>>(IN, W_e1, GM, NN);
  agg_init_k<128><<<nodeF128, B, 0, stream>>>(GM, dinv, b_e1, AG, NN);
  edge_scatter_k<128><<<edgeF128, B, 0, stream>>>(GM, src, dst, dinv, AG, NE);
  tanh_k<<<elemF128, B, 0, stream>>>(AG, (unsigned)NN * 128);

  // ---- encoder layer 2: 128 -> 128, tanh ----
  gemm_wmma<128, 128><<<gemmGrid, B, 0, stream>>>(AG, W_e2, GM, NN);
  agg_init_k<128><<<nodeF128, B, 0, stream>>>(GM, dinv, b_e2, IN, NN);
  edge_scatter_k<128><<<edgeF128, B, 0, stream>>>(GM, src, dst, dinv, IN, NE);
  tanh_k<<<elemF128, B, 0, stream>>>(IN, (unsigned)NN * 128);

  // ---- encoder layer 3 (latent): 128 -> 64, no act ----
  gemm_wmma<128, 64><<<gemmGrid, B, 0, stream>>>(IN, W_e3, GM, NN);
  agg_init_k<64><<<nodeF64, B, 0, stream>>>(GM, dinv, b_e3, AG, NN);
  edge_scatter_k<64><<<edgeF64, B, 0, stream>>>(GM, src, dst, dinv, AG, NE);

  // decoder input concat [z|cond]
  concat_zc_k<<<cdiv((long long)NN * 24, B), B, 0, stream>>>(AG, condition, IN, NN);

  // ---- decoder layer 1: 96 -> 128, tanh ----
  gemm_wmma<96, 128><<<gemmGrid, B, 0, stream>>>(IN, W_d1, GM, NN);
  agg_init_k<128><<<nodeF128, B, 0, stream>>>(GM, dinv, b_d1, AG, NN);
  edge_scatter_k<128><<<edgeF128, B, 0, stream>>>(GM, src, dst, dinv, AG, NE);
  tanh_k<<<elemF128, B, 0, stream>>>(AG, (unsigned)NN * 128);

  // ---- decoder layer 2: 128 -> 128, tanh ----
  gemm_wmma<128, 128><<<gemmGrid, B, 0, stream>>>(AG, W_d2, GM, NN);
  agg_init_k<128><<<nodeF128, B, 0, stream>>>(GM, dinv, b_d2, IN, NN);
  edge_scatter_k<128><<<edgeF128, B, 0, stream>>>(GM, src, dst, dinv, IN, NE);
  tanh_k<<<elemF128, B, 0, stream>>>(IN, (unsigned)NN * 128);

  // ---- decoder layer 3: 128 -> 128, no act, straight into d_out ----
  gemm_wmma<128, 128><<<gemmGrid, B, 0, stream>>>(IN, W_d3, GM, NN);
  agg_init_k<128><<<nodeF128, B, 0, stream>>>(GM, dinv, b_d3, OUT, NN);
  edge_scatter_k<128><<<edgeF128, B, 0, stream>>>(GM, src, dst, dinv, OUT, NE);
}